// CompletionLoss_37666863186630
// MI455X (gfx1250) — compile-verified
//
#include <hip/hip_runtime.h>
#include <math.h>

// Problem constants (match reference setup_inputs)
constexpr int T = 512;
constexpr int D = 256;
constexpr float BIG = 9999.0f;
constexpr int K = 8;

typedef __attribute__((ext_vector_type(2))) float v2f;
typedef __attribute__((ext_vector_type(8))) float v8f;

// ---------------------------------------------------------------------------
// Kernel 1: per-row stats. r[i]=sum H, q[i]=sum H^2, mse_row[i]=sum M*(X-(H-C))^2,
// and pack mask row into 8 x u32 via wave32 ballot.
// ---------------------------------------------------------------------------
__global__ void __launch_bounds__(256) row_stats_kernel(
    const float* __restrict__ X, const float* __restrict__ H,
    const float* __restrict__ C, const int* __restrict__ M,
    float* __restrict__ r, float* __restrict__ q,
    float* __restrict__ mse_row, unsigned* __restrict__ Mp) {
  const int i = blockIdx.x;
  const int d = threadIdx.x;          // 256 threads = 8 wave32
  const int idx = i * D + d;
  const float h = H[idx];
  const float x = X[idx];
  const float c = C[idx];
  const int   m = M[idx];
  const float e = (float)m * (x - (h - c));

  // pack mask bits: one 32-bit word per wave (wave32)
  unsigned long long b = __ballot(m != 0);
  if ((d & 31) == 0) Mp[i * 8 + (d >> 5)] = (unsigned)b;

  __shared__ float s_h[256], s_h2[256], s_e[256];
  s_h[d] = h; s_h2[d] = h * h; s_e[d] = e * e;
  __syncthreads();
  for (int s = 128; s > 0; s >>= 1) {
    if (d < s) {
      s_h[d]  += s_h[d + s];
      s_h2[d] += s_h2[d + s];
      s_e[d]  += s_e[d + s];
    }
    __syncthreads();
  }
  if (d == 0) { r[i] = s_h[0]; q[i] = s_h2[0]; mse_row[i] = s_e[0]; }
}

// ---------------------------------------------------------------------------
// Kernel 2: Gram matrix G = H * H^T in fp32 via V_WMMA_F32_16X16X4_F32.
// One wave per 16x16 output tile; K-loop over 256 in steps of 4.
// A-frag layout (32-bit A 16x4): lane l (half=l>>4, lr=l&15):
//   vgpr v holds A[M=lr][K = k + 2*half + v]  -> contiguous float2 per lane.
// B = H^T gives the mirrored pattern on j0. D uses the 8-VGPR C/D layout:
//   vgpr v, lane l -> D[M = v + 8*half][N = lr].
// ---------------------------------------------------------------------------
__global__ void __launch_bounds__(256) gram_wmma_kernel(
    const float* __restrict__ H, float* __restrict__ G) {
  const int lane = threadIdx.x & 31;
  const int wave = threadIdx.x >> 5;                 // 0..7
  const int i0 = (blockIdx.y * 8 + wave) * 16;       // grid.y = T/(16*8) = 4
  const int j0 = blockIdx.x * 16;                    // grid.x = T/16 = 32
  const int half = lane >> 4;
  const int lr   = lane & 15;

  const float* __restrict__ arow = H + (i0 + lr) * D + 2 * half;
  const float* __restrict__ brow = H + (j0 + lr) * D + 2 * half;

  v8f acc = {};
#pragma unroll 8
  for (int k = 0; k < D; k += 4) {
    v2f a = *(const v2f*)(arow + k);
    v2f b = *(const v2f*)(brow + k);
    // (neg_a, A, neg_b, B, c_mod, C, reuse_a, reuse_b)
    acc = __builtin_amdgcn_wmma_f32_16x16x4_f32(false, a, false, b,
                                                (short)0, acc, false, false);
  }

#pragma unroll
  for (int v = 0; v < 8; ++v) {
    G[(i0 + v + 8 * half) * T + j0 + lr] = acc[v];
  }
}

// ---------------------------------------------------------------------------
// Kernel 3: per-row scores, top-8-smallest (tie-break: lower index, matching
// jax.lax.top_k on negated scores), softmax(-topk), weighted norm sum.
// One 256-thread block per row i.
//   score[i,j] = valid ? sqrt((s2 - s1^2/256)/255) : BIG
//   s2 = q_i + q_j - 2 G_ij ; s1 = r_i - r_j
//   norm_k = sqrt(max(q_i + q_j - 2 G_ij, 0))
// ---------------------------------------------------------------------------
__global__ void __launch_bounds__(256) topk_row_kernel(
    const float* __restrict__ G, const float* __restrict__ r,
    const float* __restrict__ q, const unsigned* __restrict__ Mp,
    float* __restrict__ rowloss) {
  const int i = blockIdx.x;
  const int t = threadIdx.x;

  __shared__ float sc[T];
  __shared__ float rv[256];
  __shared__ int   ri[256];
  __shared__ unsigned mi[8];
  __shared__ float s_topv[K];
  __shared__ int   s_topi[K];

  if (t < 8) mi[t] = Mp[i * 8 + t];
  __syncthreads();

  const float qi  = q[i];
  const float rri = r[i];

  for (int j = t; j < T; j += 256) {
    bool diff = false;
#pragma unroll
    for (int w = 0; w < 8; ++w) diff |= (mi[w] != Mp[j * 8 + w]);
    const bool valid = (j != i) && diff;
    const float s2 = qi + q[j] - 2.0f * G[i * T + j];
    const float s1 = rri - r[j];
    const float var = (s2 - s1 * s1 * (1.0f / 256.0f)) * (1.0f / 255.0f);
    sc[j] = valid ? sqrtf(var) : BIG;
  }
  __syncthreads();

  // 8 rounds of argmin reduction with lower-index tie-break
  for (int kk = 0; kk < K; ++kk) {
    float v0 = sc[t]; int ix = t;
    const float v1 = sc[t + 256];
    if (v1 < v0) { v0 = v1; ix = t + 256; }   // strict < : lower index wins ties
    rv[t] = v0; ri[t] = ix;
    __syncthreads();
    for (int s = 128; s > 0; s >>= 1) {
      if (t < s) {
        const float a = rv[t], b = rv[t + s];
        if (b < a || (b == a && ri[t + s] < ri[t])) { rv[t] = b; ri[t] = ri[t + s]; }
      }
      __syncthreads();
    }
    if (t == 0) {
      s_topv[kk] = rv[0];
      s_topi[kk] = ri[0];
      sc[ri[0]] = 3.4e38f;   // exclude from later rounds
    }
    __syncthreads();
  }

  if (t == 0) {
    // softmax over -topk ; topv ascending => max(-topv) = -topv[0]
    const float mx = -s_topv[0];
    float e[K];
    float Z = 0.0f;
#pragma unroll
    for (int kk = 0; kk < K; ++kk) { e[kk] = expf(-s_topv[kk] - mx); Z += e[kk]; }
    float acc = 0.0f;
#pragma unroll
    for (int kk = 0; kk < K; ++kk) {
      const int j = s_topi[kk];
      const float d2 = qi + q[j] - 2.0f * G[i * T + j];
      acc += (e[kk] / Z) * sqrtf(fmaxf(d2, 0.0f));
    }
    rowloss[i] = acc;
  }
}

// ---------------------------------------------------------------------------
// Kernel 4: deterministic final reduction: out = sum(mse_row) + 1.0*sum(rowloss)
// ---------------------------------------------------------------------------
__global__ void __launch_bounds__(512) final_reduce_kernel(
    const float* __restrict__ mse_row, const float* __restrict__ rowloss,
    float* __restrict__ out) {
  __shared__ float s[512];
  const int t = threadIdx.x;
  s[t] = mse_row[t] + rowloss[t];   // ROW_PENALTY = 1.0
  __syncthreads();
  for (int st = 256; st > 0; st >>= 1) {
    if (t < st) s[t] += s[t + st];
    __syncthreads();
  }
  if (t == 0) out[0] = s[0];
}

// ---------------------------------------------------------------------------
// Launch
// ---------------------------------------------------------------------------
extern "C" void kernel_launch(void* const* d_in, const int* in_sizes, int n_in,
                              void* d_out, int out_size, void* d_ws, size_t ws_size,
                              hipStream_t stream) {
  const float* X = (const float*)d_in[0];
  const float* H = (const float*)d_in[1];
  const float* C = (const float*)d_in[2];
  const int*   M = (const int*)d_in[3];
  float* out = (float*)d_out;

  // Workspace layout (floats): r[512] | q[512] | mse_row[512] | rowloss[512]
  //                            | Mp[512*8 u32] | G[512*512]
  float* ws_f      = (float*)d_ws;
  float* r         = ws_f;
  float* q         = ws_f + T;
  float* mse_row   = ws_f + 2 * T;
  float* rowloss   = ws_f + 3 * T;
  unsigned* Mp     = (unsigned*)(ws_f + 4 * T);
  float* G         = ws_f + 4 * T + 8 * T;   // ws_f + 6144; G uses 262144 floats

  row_stats_kernel<<<T, 256, 0, stream>>>(X, H, C, M, r, q, mse_row, Mp);
  gram_wmma_kernel<<<dim3(T / 16, T / (16 * 8)), 256, 0, stream>>>(H, G);
  topk_row_kernel<<<T, 256, 0, stream>>>(G, r, q, Mp, rowloss);
  final_reduce_kernel<<<1, 512, 0, stream>>>(mse_row, rowloss, out);
}